// GAT_12025908429353
// MI455X (gfx1250) — compile-verified
//
#include <hip/hip_runtime.h>
#include <hip/hip_bf16.h>

// ---------------------------------------------------------------------------
// GAT (2-layer) for MI455X / gfx1250, wave32.
//   Layer 1: heads=8, ch=32  (feat 256)
//   Layer 2: heads=1, ch=64  (feat 64)
// GEMMs use V_WMMA_F32_16X16X4_F32 (full fp32, K=4 per op).
// Edge softmax: 3 atomic passes; aggregation: one wave per edge.
// ---------------------------------------------------------------------------

typedef float v2f __attribute__((ext_vector_type(2)));
typedef float v8f __attribute__((ext_vector_type(8)));

#define NEG_SLOPE 0.2f

// ---- order-preserving float <-> uint encoding for atomicMax ----------------
__device__ __forceinline__ unsigned f2ord(float f) {
  unsigned u = __float_as_uint(f);
  return (f >= 0.0f) ? (u | 0x80000000u) : ~u;
}
__device__ __forceinline__ float ord2f(unsigned u) {
  return (u & 0x80000000u) ? __uint_as_float(u & 0x7FFFFFFFu)
                           : __uint_as_float(~u);
}

// ---------------------------------------------------------------------------
// fp32 GEMM via WMMA f32 16x16x4.  C[M,NOUT] = A[M,K] @ B[K,NOUT].
// One wave computes one 16x16 output tile, looping K in steps of 4.
// A-frag layout (ISA 7.12.2): lanes 0-15 -> K=k,k+1 ; lanes 16-31 -> K=k+2,k+3
// C/D layout: VGPR r -> row r (lanes 0-15), row r+8 (lanes 16-31), col = lane&15
// ---------------------------------------------------------------------------
template <int K, int NOUT>
__global__ void gemm_wmma_f32(const float* __restrict__ A,
                              const float* __restrict__ B,
                              float* __restrict__ C, int M) {
  const int NT    = NOUT / 16;
  const int wave  = (blockIdx.x * blockDim.x + threadIdx.x) >> 5;
  const int lane  = threadIdx.x & 31;
  const int mt    = wave / NT;
  const int nt    = wave % NT;
  if (mt * 16 >= M) return;                 // wave-uniform exit (EXEC stays all-1)

  const int half = lane >> 4;               // 0: lanes 0-15, 1: lanes 16-31
  const int lr   = lane & 15;
  int arow = mt * 16 + lr;
  if (arow >= M) arow = M - 1;              // clamp (keeps EXEC all ones)
  const int bcol = nt * 16 + lr;

  v8f acc = {};
  const float* aptr = A + (size_t)arow * K + 2 * half;
  const float* bptr = B + (size_t)(2 * half) * NOUT + bcol;
  for (int k = 0; k < K; k += 4) {
    v2f a = *(const v2f*)(aptr + k);        // A[arow][k+2h], A[arow][k+2h+1]
    v2f b;
    b.x = bptr[(size_t)k * NOUT];           // B[k+2h  ][bcol]
    b.y = bptr[(size_t)(k + 1) * NOUT];     // B[k+2h+1][bcol]
    acc = __builtin_amdgcn_wmma_f32_16x16x4_f32(
        /*neg_a=*/false, a, /*neg_b=*/false, b,
        /*c_mod=*/(short)0, acc, /*reuse_a=*/false, /*reuse_b=*/false);
  }

  const int orow = mt * 16 + half * 8;
#pragma unroll
  for (int r = 0; r < 8; ++r) {
    if (orow + r < M) C[(size_t)(orow + r) * NOUT + bcol] = acc[r];
  }
}

// ---------------------------------------------------------------------------
// Per-node attention scores: a[n,h] = dot(h[n, h*ch : (h+1)*ch], att[h,:])
// ---------------------------------------------------------------------------
__global__ void att_scores(const float* __restrict__ h,
                           const float* __restrict__ att_src,
                           const float* __restrict__ att_dst,
                           float* __restrict__ asrc, float* __restrict__ adst,
                           int n, int heads, int ch) {
  int i = blockIdx.x * blockDim.x + threadIdx.x;
  if (i >= n * heads) return;
  int node = i / heads, hd = i % heads;
  const float* hp = h + (size_t)node * heads * ch + hd * ch;
  const float* as = att_src + hd * ch;
  const float* ad = att_dst + hd * ch;
  float ss = 0.0f, sd = 0.0f;
  for (int c = 0; c < ch; ++c) {
    float v = hp[c];
    ss += v * as[c];
    sd += v * ad[c];
  }
  asrc[i] = ss;
  adst[i] = sd;
}

// ---------------------------------------------------------------------------
// Pass 1: segment max over incoming edges (ordered-uint atomicMax).
// Edges e < E come from edge_index; e >= E are self loops (src=dst=e-E).
// ---------------------------------------------------------------------------
template <int H>
__global__ void edge_max(const int* __restrict__ ei, int E, int T,
                         const float* __restrict__ asrc,
                         const float* __restrict__ adst,
                         unsigned* __restrict__ m) {
  int e = blockIdx.x * blockDim.x + threadIdx.x;
  if (e >= T) return;
  int s = (e < E) ? ei[e] : (e - E);
  int d = (e < E) ? ei[E + e] : (e - E);
#pragma unroll
  for (int h = 0; h < H; ++h) {
    float v = asrc[s * H + h] + adst[d * H + h];
    v = (v > 0.0f) ? v : NEG_SLOPE * v;
    atomicMax(&m[d * H + h], f2ord(v));
  }
}

// Pass 2: segment sum of exp(e - m[dst]).
template <int H>
__global__ void edge_expsum(const int* __restrict__ ei, int E, int T,
                            const float* __restrict__ asrc,
                            const float* __restrict__ adst,
                            const unsigned* __restrict__ m,
                            float* __restrict__ ssum) {
  int e = blockIdx.x * blockDim.x + threadIdx.x;
  if (e >= T) return;
  int s = (e < E) ? ei[e] : (e - E);
  int d = (e < E) ? ei[E + e] : (e - E);
#pragma unroll
  for (int h = 0; h < H; ++h) {
    float v = asrc[s * H + h] + adst[d * H + h];
    v = (v > 0.0f) ? v : NEG_SLOPE * v;
    float ex = __expf(v - ord2f(m[d * H + h]));
    atomicAdd(&ssum[d * H + h], ex);
  }
}

// ---------------------------------------------------------------------------
// Pass 3: out[dst] += h[src] * alpha.  One wave32 per edge.
// Lanes 0..H-1 compute alpha_h once; broadcast via __shfl.
// Feature gathers are lane-coalesced (f = lane + 32*j).
// ---------------------------------------------------------------------------
template <int H, int C>
__global__ void edge_aggregate(const int* __restrict__ ei, int E, int T,
                               const float* __restrict__ asrc,
                               const float* __restrict__ adst,
                               const unsigned* __restrict__ m,
                               const float* __restrict__ ssum,
                               const float* __restrict__ hfeat,
                               float* __restrict__ out) {
  const int wave = (blockIdx.x * blockDim.x + threadIdx.x) >> 5;
  const int lane = threadIdx.x & 31;
  if (wave >= T) return;
  int s = (wave < E) ? ei[wave] : (wave - E);
  int d = (wave < E) ? ei[E + wave] : (wave - E);

  float alpha = 0.0f;
  if (lane < H) {
    float v = asrc[s * H + lane] + adst[d * H + lane];
    v = (v > 0.0f) ? v : NEG_SLOPE * v;
    alpha = __expf(v - ord2f(m[d * H + lane])) / (ssum[d * H + lane] + 1e-16f);
  }

  const int F = H * C;
  const float* hp = hfeat + (size_t)s * F;
  float* op = out + (size_t)d * F;
#pragma unroll
  for (int f = lane; f < F; f += 32) {
    float a = __shfl(alpha, f / C, 32);
    atomicAdd(&op[f], hp[f] * a);
  }
}

// ---------------------------------------------------------------------------
// small helpers
// ---------------------------------------------------------------------------
__global__ void init_bias(float* __restrict__ out, const float* __restrict__ bias,
                          int n, int F) {
  int i = blockIdx.x * blockDim.x + threadIdx.x;
  if (i < n * F) out[i] = bias[i % F];
}
__global__ void init_zero_u32(unsigned* __restrict__ p, int n) {
  int i = blockIdx.x * blockDim.x + threadIdx.x;
  if (i < n) p[i] = 0u;   // ordered encoding: below every finite float
}
__global__ void init_zero_f32(float* __restrict__ p, int n) {
  int i = blockIdx.x * blockDim.x + threadIdx.x;
  if (i < n) p[i] = 0.0f;
}
__global__ void relu_inplace(float* __restrict__ p, int n) {
  int i = blockIdx.x * blockDim.x + threadIdx.x;
  if (i < n) p[i] = fmaxf(p[i], 0.0f);
}

// ---------------------------------------------------------------------------
// launch
// ---------------------------------------------------------------------------
extern "C" void kernel_launch(void* const* d_in, const int* in_sizes, int n_in,
                              void* d_out, int out_size, void* d_ws, size_t ws_size,
                              hipStream_t stream) {
  const float* x        = (const float*)d_in[0];
  const int*   ei       = (const int*)d_in[1];
  const float* W1       = (const float*)d_in[2];
  const float* att_src1 = (const float*)d_in[3];
  const float* att_dst1 = (const float*)d_in[4];
  const float* b1       = (const float*)d_in[5];
  const float* W2       = (const float*)d_in[6];
  const float* att_src2 = (const float*)d_in[7];
  const float* att_dst2 = (const float*)d_in[8];
  const float* b2       = (const float*)d_in[9];

  const int D_IN = 256, F1 = 256, F2 = 64, H1 = 8, C1 = 32, H2 = 1, C2 = 64;
  const int N = in_sizes[0] / D_IN;
  const int E = in_sizes[1] / 2;
  const int T = E + N;                      // edges incl. self loops

  // ---- workspace layout (floats) ----
  float* ws = (float*)d_ws;
  float*    h1    = ws;                     // [N,256]  (reused for h2 [N,64])
  float*    agg1  = h1   + (size_t)N * F1;  // [N,256]
  float*    asrc1 = agg1 + (size_t)N * F1;  // [N,8]
  float*    adst1 = asrc1 + (size_t)N * H1; // [N,8]
  unsigned* m1    = (unsigned*)(adst1 + (size_t)N * H1); // [N,8]
  float*    s1    = (float*)(m1 + (size_t)N * H1);       // [N,8]
  float*    asrc2 = s1 + (size_t)N * H1;    // [N]
  float*    adst2 = asrc2 + N;              // [N]
  unsigned* m2    = (unsigned*)(adst2 + N); // [N]
  float*    s2    = (float*)(m2 + N);       // [N]
  float*    out   = (float*)d_out;          // [N,64]

  const int BT = 256;
  auto cdiv = [](long long a, long long b) { return (int)((a + b - 1) / b); };

  // ---- init: accumulators start at bias; softmax state zeroed ----
  init_bias<<<cdiv((long long)N * F1, BT), BT, 0, stream>>>(agg1, b1, N, F1);
  init_bias<<<cdiv((long long)N * F2, BT), BT, 0, stream>>>(out, b2, N, F2);
  init_zero_u32<<<cdiv((long long)N * H1, BT), BT, 0, stream>>>(m1, N * H1);
  init_zero_f32<<<cdiv((long long)N * H1, BT), BT, 0, stream>>>(s1, N * H1);
  init_zero_u32<<<cdiv(N, BT), BT, 0, stream>>>(m2, N);
  init_zero_f32<<<cdiv(N, BT), BT, 0, stream>>>(s2, N);

  // ================= Layer 1 =================
  {
    const int mtiles = (N + 15) / 16, ntiles = F1 / 16;
    const long long waves = (long long)mtiles * ntiles;
    gemm_wmma_f32<256, 256><<<cdiv(waves * 32, BT), BT, 0, stream>>>(x, W1, h1, N);
  }
  att_scores<<<cdiv((long long)N * H1, BT), BT, 0, stream>>>(
      h1, att_src1, att_dst1, asrc1, adst1, N, H1, C1);
  edge_max<8><<<cdiv(T, BT), BT, 0, stream>>>(ei, E, T, asrc1, adst1, m1);
  edge_expsum<8><<<cdiv(T, BT), BT, 0, stream>>>(ei, E, T, asrc1, adst1, m1, s1);
  edge_aggregate<8, 32><<<cdiv((long long)T * 32, BT), BT, 0, stream>>>(
      ei, E, T, asrc1, adst1, m1, s1, h1, agg1);
  relu_inplace<<<cdiv((long long)N * F1, BT), BT, 0, stream>>>(agg1, N * F1);

  // ================= Layer 2 =================
  float* h2 = h1;  // reuse h1 buffer ([N,64] fits in [N,256])
  {
    const int mtiles = (N + 15) / 16, ntiles = F2 / 16;
    const long long waves = (long long)mtiles * ntiles;
    gemm_wmma_f32<256, 64><<<cdiv(waves * 32, BT), BT, 0, stream>>>(agg1, W2, h2, N);
  }
  att_scores<<<cdiv(N, BT), BT, 0, stream>>>(
      h2, att_src2, att_dst2, asrc2, adst2, N, H2, C2);
  edge_max<1><<<cdiv(T, BT), BT, 0, stream>>>(ei, E, T, asrc2, adst2, m2);
  edge_expsum<1><<<cdiv(T, BT), BT, 0, stream>>>(ei, E, T, asrc2, adst2, m2, s2);
  edge_aggregate<1, 64><<<cdiv((long long)T * 32, BT), BT, 0, stream>>>(
      ei, E, T, asrc2, adst2, m2, s2, h2, out);
}